// harmonicLoss_36498632082236
// MI455X (gfx1250) — compile-verified
//
#include <hip/hip_runtime.h>
#include <math.h>

#define VOCAB 516
#define BDIM  32
#define SDIM  2048
#define NROWS (BDIM * VOCAB)   /* 16512 rows of length S   */
#define NCOLS (BDIM * SDIM)    /* 65536 columns of length V */
#define K2_GRID (NCOLS / 256)  /* 256 blocks                */

typedef float v2f __attribute__((ext_vector_type(2)));
typedef float v8f __attribute__((ext_vector_type(8)));

// ---------------------------------------------------------------------------
// Kernel 1: per-(b,v) row logsumexp over the sequence axis (length 2048).
// One 256-thread block per row; each thread owns 8 contiguous floats (2x b128).
// ---------------------------------------------------------------------------
__global__ void rowLseKernel(const float* __restrict__ out,
                             float* __restrict__ logZ) {
    __shared__ float red[256];
    const int row = blockIdx.x;
    const int tid = threadIdx.x;
    const float* p = out + (size_t)row * SDIM + tid * 8;
    float4 a = *(const float4*)(p);
    float4 b = *(const float4*)(p + 4);

    float m = fmaxf(fmaxf(fmaxf(a.x, a.y), fmaxf(a.z, a.w)),
                    fmaxf(fmaxf(b.x, b.y), fmaxf(b.z, b.w)));
    red[tid] = m;
    __syncthreads();
    for (int off = 128; off > 0; off >>= 1) {
        if (tid < off) red[tid] = fmaxf(red[tid], red[tid + off]);
        __syncthreads();
    }
    const float bm = red[0];
    __syncthreads();

    float s = __expf(a.x - bm) + __expf(a.y - bm) + __expf(a.z - bm) + __expf(a.w - bm)
            + __expf(b.x - bm) + __expf(b.y - bm) + __expf(b.z - bm) + __expf(b.w - bm);
    red[tid] = s;
    __syncthreads();
    for (int off = 128; off > 0; off >>= 1) {
        if (tid < off) red[tid] += red[tid + off];
        __syncthreads();
    }
    if (tid == 0) logZ[row] = bm + __logf(red[0]);
}

// ---------------------------------------------------------------------------
// Kernel 2: per-(b,s) column pass. Online logsumexp over V (for NLL) fused
// with argmax of (val - logZ_seq[v]) (for pred) and the penalty-mask lookup.
// 256 threads/block over consecutive s -> every v-step is a coalesced 128B
// load per wave. Block writes two deterministic partial sums.
// ---------------------------------------------------------------------------
__device__ __forceinline__ void lse_arg_step(float val, float lz, int v,
                                             float& m, float& sum,
                                             float& best, int& bestv) {
    if (val <= m) {
        sum += __expf(val - m);
    } else {
        sum = sum * __expf(m - val) + 1.0f;
        m = val;
    }
    const float score = val - lz;
    if (score > best) { best = score; bestv = v; }
}

__global__ void colLossKernel(const float* __restrict__ out,
                              const int* __restrict__ target,
                              const int* __restrict__ tokType,
                              const float* __restrict__ tokVal,
                              const float* __restrict__ coeff,
                              const float* __restrict__ harmonic,
                              const float* __restrict__ logZ,
                              float* __restrict__ partials) {
    __shared__ float s_logZ[VOCAB];
    __shared__ float s_tv[VOCAB];
    __shared__ int   s_tt[VOCAB];
    __shared__ float red[256];

    const int tid = threadIdx.x;
    const int col = blockIdx.x * 256 + tid;   // block spans one b (2048 % 256 == 0)
    const int bb  = col >> 11;
    const int ss  = col & (SDIM - 1);

    for (int i = tid; i < VOCAB; i += 256) {
        s_logZ[i] = logZ[bb * VOCAB + i];
        s_tv[i]   = tokVal[i];
        s_tt[i]   = tokType[i];
    }
    __syncthreads();

    const float c0 = coeff[0], c1 = coeff[1], c2 = coeff[2], c3 = coeff[3];

    const float* base = out + (size_t)bb * VOCAB * SDIM + ss;
    float m = -INFINITY, sum = 0.0f;
    float best = -INFINITY;
    int bestv = 0;

    for (int v = 0; v < VOCAB; v += 4) {        // 516 % 4 == 0; 4 loads in flight
        const float v0 = base[(size_t)(v + 0) * SDIM];
        const float v1 = base[(size_t)(v + 1) * SDIM];
        const float v2 = base[(size_t)(v + 2) * SDIM];
        const float v3 = base[(size_t)(v + 3) * SDIM];
        lse_arg_step(v0, s_logZ[v + 0], v + 0, m, sum, best, bestv);
        lse_arg_step(v1, s_logZ[v + 1], v + 1, m, sum, best, bestv);
        lse_arg_step(v2, s_logZ[v + 2], v + 2, m, sum, best, bestv);
        lse_arg_step(v3, s_logZ[v + 3], v + 3, m, sum, best, bestv);
    }

    const int   tgt     = target[col];
    const float tgt_val = base[(size_t)tgt * SDIM];
    const float nll     = (m + __logf(sum)) - tgt_val;

    // penalty mask
    const int   pt = s_tt[bestv];
    const int   gt = s_tt[tgt];
    const float d  = fabsf(s_tv[bestv] - s_tv[tgt]);
    const float pitch_w = (d == 7.0f) ? harmonic[0]
                        : (d == 5.0f) ? harmonic[1]
                        : (d == 3.0f) ? harmonic[2]
                        : (d == 4.0f) ? harmonic[3]
                        : (d == 1.0f) ? harmonic[4]
                        : (d == 2.0f) ? harmonic[5]
                        :               harmonic[6];
    const float same_w = (pt == 0) ? pitch_w
                       : (pt == 1) ? c1 * d * (1.0f / 160.0f)
                       : (pt == 2) ? c2 * d * (1.0f / 100.0f)
                       :             c3 * d * (1.0f / 128.0f);
    const float mask = (pt != gt) ? c0 : same_w;

    // deterministic block reductions
    red[tid] = nll;
    __syncthreads();
    for (int off = 128; off > 0; off >>= 1) {
        if (tid < off) red[tid] += red[tid + off];
        __syncthreads();
    }
    const float nll_sum = red[0];
    __syncthreads();

    red[tid] = mask;
    __syncthreads();
    for (int off = 128; off > 0; off >>= 1) {
        if (tid < off) red[tid] += red[tid + off];
        __syncthreads();
    }
    if (tid == 0) {
        partials[blockIdx.x]       = nll_sum;
        partials[256 + blockIdx.x] = red[0];
    }
}

// ---------------------------------------------------------------------------
// Kernel 3: single full wave (EXEC all-1s, as WMMA requires). Each lane sums
// 8 partials; the 32 lane-sums are reduced with V_WMMA_F32_16X16X4_F32 against
// an all-ones B matrix (layout-independent: each lane's value occupies exactly
// one A slot, other A VGPR = 0, C = 0 -> column-sum of D == total).
// ---------------------------------------------------------------------------
__device__ __forceinline__ float wave_sum_wmma(float laneVal) {
    v2f a;    a.x = laneVal; a.y = 0.0f;
    v2f ones; ones.x = 1.0f; ones.y = 1.0f;
    v8f c = {};
    v8f d = __builtin_amdgcn_wmma_f32_16x16x4_f32(
        false, a, false, ones, (short)0, c, false, false);
    float s = d[0] + d[1] + d[2] + d[3] + d[4] + d[5] + d[6] + d[7];
    s += __shfl_xor(s, 16, 32);   // lanes n and n+16 hold complementary M halves
    return s;
}

__global__ void finalReduceKernel(const float* __restrict__ partials,
                                  float* __restrict__ result) {
    const int lane = threadIdx.x;   // 0..31, one full wave
    float sN = 0.0f, sM = 0.0f;
    for (int i = lane; i < 256; i += 32) sN += partials[i];
    for (int i = lane; i < 256; i += 32) sM += partials[256 + i];

    const float nll_total  = wave_sum_wmma(sN);
    const float mask_total = wave_sum_wmma(sM);

    if (lane == 0) {
        const float inv = 1.0f / (float)NCOLS;
        const float loss      = nll_total * inv;
        const float mean_mask = mask_total * inv;
        result[0] = loss * (1.0f + mean_mask);   // loss + loss*mean(mask)
    }
}

// ---------------------------------------------------------------------------
extern "C" void kernel_launch(void* const* d_in, const int* in_sizes, int n_in,
                              void* d_out, int out_size, void* d_ws, size_t ws_size,
                              hipStream_t stream) {
    const float* out   = (const float*)d_in[0];   // [32,516,2048] f32
    const int*   tgt   = (const int*)  d_in[1];   // [32,2048] i32
    const int*   tt    = (const int*)  d_in[2];   // [516] i32
    const float* tv    = (const float*)d_in[3];   // [516] f32
    const float* coeff = (const float*)d_in[4];   // [4] f32
    const float* harm  = (const float*)d_in[5];   // [7] f32

    float* logZ     = (float*)d_ws;               // 16512 floats
    float* partials = logZ + NROWS;               // 512 floats

    rowLseKernel<<<NROWS, 256, 0, stream>>>(out, logZ);
    colLossKernel<<<K2_GRID, 256, 0, stream>>>(out, tgt, tt, tv, coeff, harm,
                                               logZ, partials);
    finalReduceKernel<<<1, 32, 0, stream>>>(partials, (float*)d_out);
}